// Shap_part_1_51977694216601
// MI455X (gfx1250) — compile-verified
//
#include <hip/hip_runtime.h>
#include <hip/hip_bf16.h>
#include <stdint.h>

// Problem constants (match reference)
#define N_TRAIN 1024
#define D_FEAT  128
#define N_COAL  64
#define N_INST  8
#define D4      (D_FEAT / 4)      // 32 float4 per row
#define TILE_N  64                // X_train rows staged in LDS per block (32 KB)
#define N_TILES (N_TRAIN / TILE_N)        // 16
#define MC_TOTAL (N_INST * N_COAL)        // 512 (m,c) pairs
#define MC_CHUNK 8                        // pairs handled per block
#define MC_CHUNKS (MC_TOTAL / MC_CHUNK)   // 64
// grid = N_TILES * MC_CHUNKS = 1024 blocks, 256 threads each

// Native Clang vectors (HIP's float4 is a class and is rejected by
// __builtin_nontemporal_store / the async-LDS builtins).
typedef float f4  __attribute__((ext_vector_type(4)));   // data vector, .xyzw works
typedef int   v4i __attribute__((vector_size(16)));      // matches builtin param type
typedef __attribute__((address_space(1))) v4i* gv4i_ptr; // global int4*
typedef __attribute__((address_space(3))) v4i* lv4i_ptr; // LDS int4*

#if defined(__has_builtin)
#  if __has_builtin(__builtin_amdgcn_global_load_async_to_lds_b128)
#    define HAVE_ASYNC_LDS 1
#  endif
#  if __has_builtin(__builtin_amdgcn_s_wait_asynccnt)
#    define HAVE_WAIT_ASYNC_BUILTIN 1
#  endif
#endif

__device__ __forceinline__ void async_copy_b128(const f4* gsrc, f4* ldst) {
#ifdef HAVE_ASYNC_LDS
    // Async DMA: global -> LDS, 16B per lane, tracked with ASYNCcnt.
    __builtin_amdgcn_global_load_async_to_lds_b128(
        (gv4i_ptr)(uintptr_t)gsrc, (lv4i_ptr)(uintptr_t)ldst, 0, 0);
#else
    *ldst = *gsrc;   // fallback: global_load_b128 + ds_store_b128
#endif
}

__device__ __forceinline__ void wait_async_then_barrier() {
#ifdef HAVE_ASYNC_LDS
#  ifdef HAVE_WAIT_ASYNC_BUILTIN
    __builtin_amdgcn_s_wait_asynccnt(0);
#  else
    asm volatile("s_wait_asynccnt 0" ::: "memory");
#  endif
#endif
    __syncthreads();
}

// Main kernel: out[mc][n][:] = select(mask[c], instance[m], X_train[n])
// Row layout of filled_X: row index r = mc*N_TRAIN + n, mc = m*N_COAL + c.
__global__ void __launch_bounds__(256)
shap_fill_kernel(const f4* __restrict__ X4,     // X_train as f4, [N_TRAIN*D4]
                 const f4* __restrict__ mask4,  // coalition as f4, [N_COAL*D4]
                 const f4* __restrict__ inst4,  // instance as f4, [N_INST*D4]
                 f4* __restrict__ out4)         // output as f4
{
    __shared__ f4 ldsX[TILE_N * D4];            // 32 KB

    const int bid     = blockIdx.x;
    const int tileId  = bid & (N_TILES - 1);    // n-tile
    const int mcBase  = (bid >> 4) * MC_CHUNK;  // first (m,c) pair
    const int tid     = threadIdx.x;

    // ---- Stage X_train tile into LDS with async copies (once per block) ----
    {
        const f4* gbase = X4 + (size_t)tileId * (TILE_N * D4);
        #pragma unroll
        for (int k = 0; k < (TILE_N * D4) / 256; ++k) {      // 8 iterations
            const int idx = k * 256 + tid;
            async_copy_b128(gbase + idx, &ldsX[idx]);
        }
    }
    wait_async_then_barrier();

    const int lane     = tid & 31;   // which f4 of the 128-float row
    const int rowInBlk = tid >> 5;   // 0..7: row phase within tile

    // ---- Select + non-temporal streaming stores ----
    #pragma unroll
    for (int p = 0; p < MC_CHUNK; ++p) {
        const int mc = mcBase + p;
        const int m  = mc >> 6;          // / N_COAL
        const int c  = mc & (N_COAL - 1);

        // loop-invariant per (m,c): held in registers, compares hoisted
        const f4 mk = mask4[c * D4 + lane];
        const f4 iv = inst4[m * D4 + lane];
        const bool bx = (mk.x != 0.0f);
        const bool by = (mk.y != 0.0f);
        const bool bz = (mk.z != 0.0f);
        const bool bw = (mk.w != 0.0f);

        const size_t rowBase = (size_t)mc * N_TRAIN + (size_t)tileId * TILE_N;
        f4* __restrict__ obase = out4 + rowBase * D4 + lane;

        #pragma unroll
        for (int n0 = rowInBlk; n0 < TILE_N; n0 += 8) {
            const f4 x = ldsX[n0 * D4 + lane];   // ds_load_b128
            f4 r;
            r.x = bx ? iv.x : x.x;                // v_cndmask_b32 x4
            r.y = by ? iv.y : x.y;
            r.z = bz ? iv.z : x.z;
            r.w = bw ? iv.w : x.w;
            __builtin_nontemporal_store(r, obase + (size_t)n0 * D4);
        }
    }
}

// Tail: filled_X rows [M*C*N, M*C*N+M) = instance, then tuple element 2 = instance.
__global__ void __launch_bounds__(256)
shap_tail_kernel(const f4* __restrict__ inst4, f4* __restrict__ out4)
{
    const int tid = threadIdx.x;                 // 256 threads = 256 f4 = 1024 floats
    const f4 v = inst4[tid];
    const size_t filledRows = (size_t)MC_TOTAL * N_TRAIN + N_INST;   // 524296
    f4* t1 = out4 + (size_t)MC_TOTAL * N_TRAIN * D4;                 // appended instance rows
    f4* t2 = out4 + filledRows * D4;                                 // second tuple output
    __builtin_nontemporal_store(v, t1 + tid);
    __builtin_nontemporal_store(v, t2 + tid);
}

extern "C" void kernel_launch(void* const* d_in, const int* in_sizes, int n_in,
                              void* d_out, int out_size, void* d_ws, size_t ws_size,
                              hipStream_t stream) {
    (void)in_sizes; (void)n_in; (void)d_ws; (void)ws_size; (void)out_size;
    const f4* X4    = (const f4*)d_in[0];   // X_train [1024,128] f32
    const f4* mask4 = (const f4*)d_in[1];   // coalition_vectors [64,128] f32 (0/1)
    const f4* inst4 = (const f4*)d_in[2];   // instance [8,128] f32
    f4* out4 = (f4*)d_out;

    shap_fill_kernel<<<N_TILES * MC_CHUNKS, 256, 0, stream>>>(X4, mask4, inst4, out4);
    shap_tail_kernel<<<1, 256, 0, stream>>>(inst4, out4);
}